// Projection_Element_68994354643034
// MI455X (gfx1250) — compile-verified
//
#include <hip/hip_runtime.h>

// Streaming L2-ball projection: y = x / max(1, ||x||_2) for 16.7M 2-D points.
// HBM-bound (AI ~0.4 FLOP/B). MI455X strategy:
//   * hot kernel has NO bounds checks (full blocks only) -> no exec-mask
//     gymnastics / zero-fill v_movs; remainder handled by separate kernels
//   * 4x b128 loads/stores per thread with immediate offsets (64 B/thread each way)
//   * scale = rsqrt(max(n2,1)) via v_rsq_f32 + 1 Newton step (~1-2 ulp;
//     interior points exact since rsq(1)=1 is an NR fixed point)
//   * RT loads keep the 128 MiB input L2-resident (192 MB L2) across graph
//     replays; NT stores keep the write stream from evicting it.

typedef float __attribute__((ext_vector_type(4))) v4f;

#define TPB   256
#define CHUNK 4  // float4s per thread -> 1024 float4s per block

__device__ __forceinline__ v4f project2(v4f v) {
    // point 0 = (v.x,v.y), point 1 = (v.z,v.w)
    float n2a = __builtin_fmaf(v.x, v.x, v.y * v.y);
    float n2b = __builtin_fmaf(v.z, v.z, v.w * v.w);
    float ma  = __builtin_fmaxf(n2a, 1.0f);
    float mb  = __builtin_fmaxf(n2b, 1.0f);

    float ra = __builtin_amdgcn_rsqf(ma);
    float rb = __builtin_amdgcn_rsqf(mb);
    float ta = 0.5f * ma * ra;
    float tb = 0.5f * mb * rb;
    ra = ra * __builtin_fmaf(-ta, ra, 1.5f);
    rb = rb * __builtin_fmaf(-tb, rb, 1.5f);

    v4f o;
    o.x = v.x * ra;
    o.y = v.y * ra;
    o.z = v.z * rb;
    o.w = v.w * rb;
    return o;
}

// Hot path: every block is full, no bounds checks anywhere.
__global__ __launch_bounds__(TPB) void proj_l2ball_full(const v4f* __restrict__ x,
                                                        v4f* __restrict__ y) {
    const int base = blockIdx.x * (TPB * CHUNK) + threadIdx.x;
    const v4f* __restrict__ xb = x + base;
    v4f*       __restrict__ yb = y + base;

    v4f v[CHUNK];
#pragma unroll
    for (int k = 0; k < CHUNK; ++k)
        v[k] = xb[k * TPB];  // b128 loads, byte offset k*4096 folded into insn

#pragma unroll
    for (int k = 0; k < CHUNK; ++k)
        __builtin_nontemporal_store(project2(v[k]), &yb[k * TPB]);  // b128 NT
}

// Remainder float4s (guarded, 1 per thread).
__global__ __launch_bounds__(TPB) void proj_l2ball_rem(const v4f* __restrict__ x,
                                                       v4f* __restrict__ y,
                                                       int start4, int n4) {
    int i = start4 + blockIdx.x * blockDim.x + threadIdx.x;
    if (i >= n4) return;
    v4f v = x[i];
    __builtin_nontemporal_store(project2(v), &y[i]);
}

// Trailing point when flat float count % 4 == 2.
__global__ __launch_bounds__(64) void proj_l2ball_tail(const float* __restrict__ x,
                                                       float* __restrict__ y,
                                                       int first_pair,
                                                       int n_pairs) {
    int p = first_pair + blockIdx.x * blockDim.x + threadIdx.x;
    if (p >= n_pairs) return;
    float a = x[2 * p + 0];
    float b = x[2 * p + 1];
    float n2 = __builtin_fmaf(a, a, b * b);
    float m  = __builtin_fmaxf(n2, 1.0f);
    float r  = __builtin_amdgcn_rsqf(m);
    float t  = 0.5f * m * r;
    r = r * __builtin_fmaf(-t, r, 1.5f);
    y[2 * p + 0] = a * r;
    y[2 * p + 1] = b * r;
}

extern "C" void kernel_launch(void* const* d_in, const int* in_sizes, int n_in,
                              void* d_out, int out_size, void* d_ws, size_t ws_size,
                              hipStream_t stream) {
    (void)n_in; (void)d_ws; (void)ws_size; (void)out_size;

    const float* x = (const float*)d_in[0];
    float*       y = (float*)d_out;

    const int n  = in_sizes[0];          // flat float count (B*2 = 33,554,432)
    const int n4 = n / 4;                // float4 chunks (2 points each)

    const int per_block   = TPB * CHUNK; // 1024 float4s per block
    const int full_blocks = n4 / per_block;
    const int start_rem   = full_blocks * per_block;

    if (full_blocks > 0) {
        proj_l2ball_full<<<full_blocks, TPB, 0, stream>>>((const v4f*)x, (v4f*)y);
    }
    if (start_rem < n4) {
        int rem = n4 - start_rem;
        int blocks = (rem + TPB - 1) / TPB;
        proj_l2ball_rem<<<blocks, TPB, 0, stream>>>((const v4f*)x, (v4f*)y,
                                                    start_rem, n4);
    }

    // Trailing pair if flat count % 4 == 2 (n is always even for 2-D points).
    const int n_pairs    = n / 2;
    const int first_pair = (n4 * 4) / 2;
    if (first_pair < n_pairs) {
        int tail = n_pairs - first_pair;
        int blocks = (tail + 63) / 64;
        proj_l2ball_tail<<<blocks, 64, 0, stream>>>(x, y, first_pair, n_pairs);
    }
}